// CapsuleLayer_28544352649934
// MI455X (gfx1250) — compile-verified
//
#include <hip/hip_runtime.h>
#include <hip/hip_bf16.h>
#include <math.h>

typedef float v2f __attribute__((ext_vector_type(2)));
typedef float v8f __attribute__((ext_vector_type(8)));

#define B_ 256
#define C_ 10
#define R_ 1152
#define I_ 8
#define O_ 16
#define WAVES 8
#define RPW (R_ / WAVES)   // 144 route nodes per wave
#define N_ITERS 3

// ---- DPP16 cross-lane sum over each 16-lane row (no LDS, no ds_bpermute) ----
template <int CTRL>
__device__ __forceinline__ float dpp_add(float v) {
  int s = __builtin_amdgcn_update_dpp(0, __builtin_bit_cast(int, v),
                                      CTRL, 0xF, 0xF, true);
  return v + __builtin_bit_cast(float, s);
}

__device__ __forceinline__ float row16_sum(float v) {
  v = dpp_add<0xB1>(v);   // quad_perm [1,0,3,2]  (xor 1)
  v = dpp_add<0x4E>(v);   // quad_perm [2,3,0,1]  (xor 2)
  v = dpp_add<0x141>(v);  // row_half_mirror      (pairs quads 0<->1)
  v = dpp_add<0x140>(v);  // row_mirror           (pairs 8-groups)
  return v;
}

// u_tile[16 batch x 16 out] = x_tile[16x8] * W[8x16]; K=8 via 2 chained WMMAs
__device__ __forceinline__ v8f make_u(const float* __restrict__ xr,
                                      const float* __restrict__ wr,
                                      int half, int col) {
  v8f u = {};
  #pragma unroll
  for (int chunk = 0; chunk < 2; ++chunk) {
    const int kb = 4 * chunk + 2 * half;
    v2f a;                                  // A: K = kb, kb+1 for this lane
    a.x = xr[kb + 0];
    a.y = xr[kb + 1];
    v2f bm;                                 // B: VGPR j -> K = kb + j, N = col
    bm.x = wr[(kb + 0) * O_ + col];
    bm.y = wr[(kb + 1) * O_ + col];
    u = __builtin_amdgcn_wmma_f32_16x16x4_f32(
            false, a, false, bm, (short)0, u, false, false);
  }
  return u;
}

// One workgroup = one (capsule c, 16-batch tile). 8 waves split the R dimension.
// u is never materialized (189 MB if it were): recomputed per iteration via
// WMMA from L2-resident x / route_weights (15 MB total, L2 = 192 MB).
// Since b starts at 0 and b += u.v, logits at iter k are u . (sum of past v's):
// the only cross-iteration state is vsum[16] per (b,c); each iteration is one
// online-softmax pass over r (flash-attention style), merged across waves.
__global__ __launch_bounds__(256) void caps_route_kernel(
    const float* __restrict__ x,      // [B, R, I]
    const float* __restrict__ w,      // [C, R, I, O]
    float* __restrict__ out)          // [B, C, 1, 1, O]
{
  const int c    = blockIdx.x % C_;
  const int b0   = (blockIdx.x / C_) * 16;
  const int lane = threadIdx.x & 31;
  const int wave = threadIdx.x >> 5;
  const int col  = lane & 15;        // N / output-capsule index in C/D layout
  const int half = lane >> 4;        // row-half selector (M or K offset)

  __shared__ float ls_S[WAVES][16][16];
  __shared__ float ls_m[WAVES][16];
  __shared__ float ls_d[WAVES][16];

  // A-matrix row for this lane is batch row M = col (same M in both halves).
  const float* xrow  = x + (size_t)(b0 + col) * (R_ * I_);
  const float* wbase = w + (size_t)c * (R_ * I_ * O_);

  const int r0 = wave * RPW;

  float vs[8];      // accumulated sum of v's (routing logits = u . vs), tile layout
  float vlast[8];   // v from the most recent squash
  #pragma unroll
  for (int j = 0; j < 8; ++j) { vs[j] = 0.0f; vlast[j] = 0.0f; }

  for (int it = 0; it < N_ITERS; ++it) {
    float m[8], dn[8], S[8];

    if (it == 0) {
      // ---- iteration 0: softmax(0) is exactly uniform -> plain accumulation
      #pragma unroll
      for (int j = 0; j < 8; ++j) { m[j] = 0.0f; dn[j] = (float)RPW; S[j] = 0.0f; }
      for (int rr = 0; rr < RPW; ++rr) {
        const int r = r0 + rr;
        const float* xr = xrow  + r * I_;
        const float* wr = wbase + r * (I_ * O_);
        __builtin_prefetch(wr + 4 * I_ * O_, 0, 3);   // global_prefetch_b8
        __builtin_prefetch(xr + 4 * I_, 0, 3);
        const v8f u = make_u(xr, wr, half, col);
        #pragma unroll
        for (int j = 0; j < 8; ++j) S[j] += u[j];
      }
    } else {
      // ---- online-softmax pass over this wave's r-slice ----
      #pragma unroll
      for (int j = 0; j < 8; ++j) { m[j] = -INFINITY; dn[j] = 0.0f; S[j] = 0.0f; }
      for (int rr = 0; rr < RPW; ++rr) {
        const int r = r0 + rr;
        const float* xr = xrow  + r * I_;
        const float* wr = wbase + r * (I_ * O_);
        const v8f u = make_u(xr, wr, half, col);
        #pragma unroll
        for (int j = 0; j < 8; ++j) {
          // logit a_row = dot(u_row, vsum_row), reduced over 16 output dims
          const float p = row16_sum(u[j] * vs[j]);
          const float nm = fmaxf(m[j], p);
          const float sc = __expf(m[j] - nm);
          const float wg = __expf(p - nm);
          dn[j] = dn[j] * sc + wg;
          S[j]  = S[j]  * sc + wg * u[j];
          m[j]  = nm;
        }
      }
    }

    // ---- publish per-wave partials ----
    #pragma unroll
    for (int j = 0; j < 8; ++j)
      ls_S[wave][j + 8 * half][col] = S[j];
    if (col == 0) {
      #pragma unroll
      for (int j = 0; j < 8; ++j) {
        ls_m[wave][j + 8 * half] = m[j];
        ls_d[wave][j + 8 * half] = dn[j];
      }
    }
    __syncthreads();

    // ---- merge the 8 wave-partials (redundantly in every lane) ----
    float sv[8];
    #pragma unroll
    for (int j = 0; j < 8; ++j) {
      const int row = j + 8 * half;
      float M = -INFINITY;
      #pragma unroll
      for (int wv = 0; wv < WAVES; ++wv) M = fmaxf(M, ls_m[wv][row]);
      float D = 0.0f, Sm = 0.0f;
      #pragma unroll
      for (int wv = 0; wv < WAVES; ++wv) {
        const float e = __expf(ls_m[wv][row] - M);
        D  += ls_d[wv][row] * e;
        Sm += ls_S[wv][row][col] * e;
      }
      sv[j] = Sm / D;                            // s = softmax-weighted sum of u
    }
    __syncthreads();                             // LDS reusable next iteration

    // ---- squash: v = s * sqrt(|s|^2) / (1 + |s|^2) ----
    #pragma unroll
    for (int j = 0; j < 8; ++j) {
      const float q   = row16_sum(sv[j] * sv[j]);   // |s|^2 over output dims
      const float fac = sqrtf(q) / (1.0f + q);
      const float vj  = sv[j] * fac;
      vlast[j] = vj;
      vs[j]   += vj;                             // cumulative logit direction
    }
  }

  // one wave writes the (identical) result: out[b, c, 0, 0, o]
  if (wave == 0) {
    #pragma unroll
    for (int j = 0; j < 8; ++j) {
      const int row = j + 8 * half;
      out[((size_t)(b0 + row) * C_ + c) * O_ + col] = vlast[j];
    }
  }
}

extern "C" void kernel_launch(void* const* d_in, const int* in_sizes, int n_in,
                              void* d_out, int out_size, void* d_ws, size_t ws_size,
                              hipStream_t stream) {
  const float* x  = (const float*)d_in[0];   // [256, 1152, 8]
  const float* rw = (const float*)d_in[1];   // [10, 1152, 8, 16]
  float* out = (float*)d_out;                // [256, 10, 1, 1, 16]
  (void)in_sizes; (void)n_in; (void)out_size; (void)d_ws; (void)ws_size;

  dim3 grid((B_ / 16) * C_);                 // 160 workgroups
  dim3 block(256);                           // 8 waves (wave32)
  caps_route_kernel<<<grid, block, 0, stream>>>(x, rw, out);
}